// KernelizedHeadAttention_85194971284032
// MI455X (gfx1250) — compile-verified
//
#include <hip/hip_runtime.h>
#include <hip/hip_bf16.h>
#include <math.h>

typedef __attribute__((ext_vector_type(2))) float v2f;
typedef __attribute__((ext_vector_type(8))) float v8f;

#define NB 4
#define NS 1024
#define ND 2048
#define NH 16
#define NDH 128
#define NDHID 128
#define NDKER 64
#define FEPS 1e-6f

__device__ __forceinline__ float gelu_exact(float x) {
    // torch F.gelu default: 0.5*x*(1+erf(x/sqrt(2)))
    return 0.5f * x * (1.0f + erff(x * 0.70710678118654752f));
}

__device__ __forceinline__ v8f wmma_f32(v2f a, v2f b, v8f c) {
    // V_WMMA_F32_16X16X4_F32 : D = A(16x4) * B(4x16) + C(16x16), fp32
    return __builtin_amdgcn_wmma_f32_16x16x4_f32(
        /*neg_a=*/false, a, /*neg_b=*/false, b,
        /*c_mod=*/(short)0, c, /*reuse_a=*/false, /*reuse_b=*/false);
}

// ---- gfx1250 async global->LDS copy (ASYNCcnt-tracked) --------------------
__device__ __forceinline__ void async_copy16(void* lds, const void* g) {
    asm volatile("global_load_async_to_lds_b128 %0, %1, off"
                 :: "v"((unsigned)(uintptr_t)lds),
                    "v"((unsigned long long)(uintptr_t)g)
                 : "memory");
}

__device__ __forceinline__ void wait_async0() {
#if defined(__has_builtin) && __has_builtin(__builtin_amdgcn_s_wait_asynccnt)
    __builtin_amdgcn_s_wait_asynccnt(0);
#else
    asm volatile("s_wait_asynccnt 0x0" ::: "memory");
#endif
}

// ---------------------------------------------------------------------------
// Feature-map kernel: one wave (32 threads) per (b,h,16-row tile).
//   q-mode (isK=0): qf = |gelu(gelu(X@W1)@W2)|
//   k-mode (isK=1): t  = gelu(gelu(X@W1)@W2) * |scalingD|
//                   kf = |t + (t@interaction)*scalingD2|
// ---------------------------------------------------------------------------
__global__ __launch_bounds__(32) void feature_kernel(
    const float* __restrict__ x,    // [B,S,D]
    const float* __restrict__ w1,   // [H,128,128]
    const float* __restrict__ w2,   // [H,128,64]
    const float* __restrict__ ik,   // [H,64,64]
    const float* __restrict__ sD,   // [1,H,1,64]
    const float* __restrict__ sD2,  // [1,H,1,64]
    float* __restrict__ feat,       // [B,H,S,64]
    int isK)
{
    __shared__ float ldsX[16 * 128];
    __shared__ float ldsY[16 * 128];

    int tile  = blockIdx.x;
    int stile = tile & 63;      // S/16 = 64 tiles
    int bh    = tile >> 6;
    int b     = bh >> 4;
    int h     = bh & 15;
    int srow0 = stile * 16;

    int lane = threadIdx.x;
    int n    = lane & 15;       // row (A) / col (B,C) owned by this lane
    int hi   = lane >> 4;       // which K/M half

    // stage 16x128 input tile into LDS (coalesced float4 rows)
    {
        const float* base = x + (size_t)b * NS * ND + (size_t)h * NDH;
        for (int i = 0; i < 16; ++i) {
            const float4* src = (const float4*)(base + (size_t)(srow0 + i) * ND);
            ((float4*)ldsX)[i * 32 + lane] = src[lane];
        }
    }
    __syncthreads();

    // GEMM1: Y[16x128] = gelu(X @ W1[h]), K=128
    const float* W1 = w1 + (size_t)h * 128 * 128;
    for (int nt = 0; nt < 8; ++nt) {
        v8f acc = {};
        for (int kk = 0; kk < 32; ++kk) {
            int k0 = kk * 4 + 2 * hi;
            v2f a = *(const v2f*)&ldsX[n * 128 + k0];
            v2f bm;
            bm.x = W1[(size_t)k0 * 128 + nt * 16 + n];
            bm.y = W1[(size_t)(k0 + 1) * 128 + nt * 16 + n];
            acc = wmma_f32(a, bm, acc);
        }
        for (int i = 0; i < 8; ++i) {
            int m = i + 8 * hi;
            ldsY[m * 128 + nt * 16 + n] = gelu_exact(acc[i]);
        }
    }
    __syncthreads();

    // GEMM2: T[16x64] = Y @ W2[h], K=128
    const float* W2 = w2 + (size_t)h * 128 * 64;
    for (int nt = 0; nt < 4; ++nt) {
        v8f acc = {};
        for (int kk = 0; kk < 32; ++kk) {
            int k0 = kk * 4 + 2 * hi;
            v2f a = *(const v2f*)&ldsY[n * 128 + k0];
            v2f bm;
            bm.x = W2[(size_t)k0 * 64 + nt * 16 + n];
            bm.y = W2[(size_t)(k0 + 1) * 64 + nt * 16 + n];
            acc = wmma_f32(a, bm, acc);
        }
        if (!isK) {
            for (int i = 0; i < 8; ++i) {
                int m = i + 8 * hi;
                feat[((size_t)bh * NS + srow0 + m) * NDKER + nt * 16 + n] =
                    fabsf(gelu_exact(acc[i]));
            }
        } else {
            for (int i = 0; i < 8; ++i) {
                int m = i + 8 * hi;
                float t = gelu_exact(acc[i]) * fabsf(sD[h * 64 + nt * 16 + n]);
                ldsX[m * 64 + nt * 16 + n] = t;   // ldsX reused as T buffer
            }
        }
    }

    if (isK) {
        __syncthreads();
        // GEMM3: KF = T + (T @ interaction[h]) * scalingD2, K=64
        const float* IK = ik + (size_t)h * 64 * 64;
        for (int nt = 0; nt < 4; ++nt) {
            v8f acc = {};
            for (int kk = 0; kk < 16; ++kk) {
                int k0 = kk * 4 + 2 * hi;
                v2f a = *(const v2f*)&ldsX[n * 64 + k0];
                v2f bm;
                bm.x = IK[(size_t)k0 * 64 + nt * 16 + n];
                bm.y = IK[(size_t)(k0 + 1) * 64 + nt * 16 + n];
                acc = wmma_f32(a, bm, acc);
            }
            for (int i = 0; i < 8; ++i) {
                int m = i + 8 * hi;
                float t = ldsX[m * 64 + nt * 16 + n];
                float val = t + acc[i] * sD2[h * 64 + nt * 16 + n];
                feat[((size_t)bh * NS + srow0 + m) * NDKER + nt * 16 + n] = fabsf(val);
            }
        }
    }
}

// ---------------------------------------------------------------------------
// Single-pass attention kernel: 4 waves per block, each wave owns a 16-row
// tile of a shared 64-row strip of one (b,h). K/V tiles double-buffered in
// LDS via async global->LDS DMA (ASYNCcnt) and shared by all 4 waves.
//   p[m,t]   = mask ? score+eps : exp(sparse_w)        (unnormalized attn)
//   rowsum  += mask ? score : 0
//   accO    += p @ V            (via LDS C->A layout round trip)
// final:  out[m,:] = accO[m,:] * 1/(rowsum + eps + exp(sparse_lse))
// ---------------------------------------------------------------------------
__global__ __launch_bounds__(128) void attn_kernel(
    const float* __restrict__ qf,           // [B,H,S,64] (abs)
    const float* __restrict__ kf,           // [B,H,S,64] (abs)
    const float* __restrict__ vin,          // [B,S,D]
    const unsigned char* __restrict__ mask, // [B,1,S,S] bool
    const float* __restrict__ slse,         // [B,H,S,1]
    const float* __restrict__ sw,           // [B,H,S,S]
    float* __restrict__ out)                // [B,S,D]
{
    __shared__ float ldsK[2][16 * 64];      // kf tile double buffer, 8 KB
    __shared__ float ldsV[2][16 * 128];     // V tile double buffer, 16 KB
    __shared__ float ldsA[4][16 * 16];      // per-wave p tiles, 4 KB

    int tile  = blockIdx.x;                 // B*H*(S/64) = 1024
    int stile = tile & 15;                  // S/64 = 16 strips
    int bh    = tile >> 4;
    int b     = bh >> 4;
    int h     = bh & 15;

    int tid   = threadIdx.x;
    int wave  = tid >> 5;
    int lane  = tid & 31;
    int n     = lane & 15;
    int hi    = lane >> 4;
    int srow0 = stile * 64 + wave * 16;

    float* myA = &ldsA[wave][0];

    const float* kbase = kf + (size_t)bh * NS * NDKER;
    const float* vbase = vin + (size_t)b * NS * ND + (size_t)h * NDH;

    // Preload Qf A-fragments for all 16 K-steps (K=64)
    v2f aQ[16];
    {
        const float* qrow = qf + ((size_t)bh * NS + srow0 + n) * NDKER;
        for (int kk = 0; kk < 16; ++kk)
            aQ[kk] = *(const v2f*)&qrow[kk * 4 + 2 * hi];
    }

    const unsigned char* mrow0 = mask + (size_t)b * NS * NS + (size_t)srow0 * NS;
    const float* swrow0 = sw + ((size_t)bh * NS + srow0) * NS;

    float psum[8];
    for (int i = 0; i < 8; ++i) psum[i] = 0.0f;
    v8f accO[8] = {};

    // ---- async stage of one (K,V) tile pair into buffer `buf` ----------
    auto stage = [&](int t, int buf) {
        // K tile: 16x64 fp32, rows contiguous -> 1024 contiguous floats
        const float4* ksrc = (const float4*)(kbase + (size_t)t * 16 * NDKER);
        float4* kdst = (float4*)&ldsK[buf][0];
        async_copy16(&kdst[tid],       &ksrc[tid]);
        async_copy16(&kdst[tid + 128], &ksrc[tid + 128]);
        // V tile: 16 rows x 128 floats (row stride ND)
        float4* vdst = (float4*)&ldsV[buf][0];
        for (int i = 0; i < 4; ++i) {
            int idx = i * 128 + tid;        // float4 index 0..511
            int row = idx >> 5;
            int c4  = idx & 31;
            async_copy16(&vdst[idx],
                         (const float4*)(vbase + (size_t)(t * 16 + row) * ND + c4 * 4));
        }
    };

    stage(0, 0);
    for (int t = 0; t < 64; ++t) {
        int cur = t & 1;
        wait_async0();       // this wave's async loads into buffer `cur` done
        __syncthreads();     // all waves' loads visible; prev compute retired
        if (t + 1 < 64)
            stage(t + 1, cur ^ 1);   // overlap next-tile DMA with compute

        // ---- scores: 16x16 tile, K=64 via 16 fp32 WMMA -----------------
        v8f acc = {};
        for (int kk = 0; kk < 16; ++kk) {
            int k0 = kk * 4 + 2 * hi;
            v2f bm = *(const v2f*)&ldsK[cur][n * 64 + k0];
            acc = wmma_f32(aQ[kk], bm, acc);
        }

        // ---- unnormalized attn tile + masked row sums ------------------
        for (int i = 0; i < 8; ++i) {
            int m = i + 8 * hi;
            unsigned char mk = mrow0[(size_t)m * NS + t * 16 + n];
            float sval = acc[i];
            psum[i] += mk ? sval : 0.0f;
            // sparse weights are a read-once 256MB stream: keep them out of L2
            float swv = __builtin_nontemporal_load(&swrow0[(size_t)m * NS + t * 16 + n]);
            float p = mk ? (sval + FEPS) : __expf(swv);
            myA[m * 16 + n] = p;
        }
        __syncthreads();   // p tiles visible (C->A layout round trip)

        // ---- accO += p @ V (K=16 per t-tile, 32 fp32 WMMA) -------------
        for (int nt = 0; nt < 8; ++nt) {
            v8f a8 = accO[nt];
            for (int kk = 0; kk < 4; ++kk) {
                int k0 = kk * 4 + 2 * hi;
                v2f a = *(const v2f*)&myA[n * 16 + k0];
                v2f bm;
                bm.x = ldsV[cur][k0 * 128 + nt * 16 + n];
                bm.y = ldsV[cur][(k0 + 1) * 128 + nt * 16 + n];
                a8 = wmma_f32(a, bm, a8);
            }
            accO[nt] = a8;
        }
    }

    // ---- per-row normalization: reduce masked sums, scale accumulators --
    float invZ[8];
    for (int i = 0; i < 8; ++i) {
        float s = psum[i];
        s += __shfl_xor(s, 1, 32);
        s += __shfl_xor(s, 2, 32);
        s += __shfl_xor(s, 4, 32);
        s += __shfl_xor(s, 8, 32);   // xor masks stay within each 16-lane half
        int m = i + 8 * hi;
        float Z = s + FEPS + expf(slse[(size_t)bh * NS + srow0 + m]);
        invZ[i] = 1.0f / Z;
    }

    float* obase = out + (size_t)b * NS * ND + (size_t)h * NDH;
    for (int nt = 0; nt < 8; ++nt)
        for (int i = 0; i < 8; ++i) {
            int m = i + 8 * hi;
            __builtin_nontemporal_store(accO[nt][i] * invZ[i],
                &obase[(size_t)(srow0 + m) * ND + nt * 16 + n]);
        }
}

extern "C" void kernel_launch(void* const* d_in, const int* in_sizes, int n_in,
                              void* d_out, int out_size, void* d_ws, size_t ws_size,
                              hipStream_t stream) {
    (void)in_sizes; (void)n_in; (void)out_size; (void)ws_size;
    const float* q    = (const float*)d_in[0];
    const float* k    = (const float*)d_in[1];
    const float* v    = (const float*)d_in[2];
    const unsigned char* mask = (const unsigned char*)d_in[3];
    const float* slse = (const float*)d_in[4];
    const float* sw   = (const float*)d_in[5];
    const float* wq1  = (const float*)d_in[6];
    const float* wk1  = (const float*)d_in[7];
    const float* wq2  = (const float*)d_in[8];
    const float* wk2  = (const float*)d_in[9];
    const float* ik   = (const float*)d_in[10];
    const float* sD   = (const float*)d_in[11];
    const float* sD2  = (const float*)d_in[12];
    float* out = (float*)d_out;

    float* qf = (float*)d_ws;                                 // 16 MB
    float* kf = qf + (size_t)NB * NH * NS * NDKER;            // 16 MB

    dim3 fgrid(NB * NH * (NS / 16));   // 4096 waves
    dim3 fblk(32);
    feature_kernel<<<fgrid, fblk, 0, stream>>>(q, wq1, wq2, ik, sD, sD2, qf, 0);
    feature_kernel<<<fgrid, fblk, 0, stream>>>(k, wk1, wk2, ik, sD, sD2, kf, 1);

    dim3 agrid(NB * NH * (NS / 64));   // 1024 blocks x 4 waves
    dim3 ablk(128);
    attn_kernel<<<agrid, ablk, 0, stream>>>(qf, kf, v, mask, slse, sw, out);
}